// SimpleGRU_1340029796479
// MI455X (gfx1250) — compile-verified
//
#include <hip/hip_runtime.h>

// GRU: B=8192, T=512, I=5, H=8, OUT=2  (torch gate order r,z,n)
// Wave of 32 lanes handles 16 batch elements using V_WMMA_F32_16X16X4_F32.
// Batch b <-> lane pair (b, b+16).  Gate tiles: tile0 rows = [r0..r7, z0..z7],
// tile1 rows = [n0..n7, zero x8].
//
// A-layout (16x4 f32, 2 VGPRs): lane l<16: {A[l,k0],A[l,k0+1]}, lane l+16: {A[l-16,k0+2],A[l-16,k0+3]}
// B-layout (4x16 f32, 2 VGPRs): lane n<16: {B[k0,n],B[k0+1,n]},  lane n+16: {B[k0+2,n],B[k0+3,n]}
// D-layout (16x16 f32, 8 VGPRs): lane n<16: D[0..7,n],           lane n+16: D[8..15,n]

typedef float v2f __attribute__((ext_vector_type(2)));
typedef float v8f __attribute__((ext_vector_type(8)));

#define I_DIM 5
#define H_DIM 8
#define T_DIM 512

__device__ __forceinline__ v2f mk2(float a, float b) { v2f v; v.x = a; v.y = b; return v; }

// gfx1250 has V_TANH_F32 (TRANS32). Use it directly; both tanh and sigmoid
// become a single transcendental instead of exp + IEEE-divide expansions.
#if __has_builtin(__builtin_amdgcn_tanhf)
__device__ __forceinline__ float tanh_hw(float x) {
  return __builtin_amdgcn_tanhf(x);
}
#else
__device__ __forceinline__ float tanh_hw(float x) {
  float e = __expf(-2.0f * x);                       // v_exp_f32
  return (1.0f - e) * __builtin_amdgcn_rcpf(1.0f + e); // v_rcp_f32 (approx, no div-scale)
}
#endif
__device__ __forceinline__ float sigmoid_f(float x) {
  return __builtin_fmaf(0.5f, tanh_hw(0.5f * x), 0.5f);
}

// swap value with lane^16 partner (wave32): group-of-32 swizzle, xor=0x10,and=0x1f
__device__ __forceinline__ float swz16(float v) {
  return __int_as_float(__builtin_amdgcn_ds_swizzle(__float_as_int(v), 0x401F));
}

__device__ __forceinline__ v8f wmma4(v2f a, v2f b, v8f c) {
  return __builtin_amdgcn_wmma_f32_16x16x4_f32(false, a, false, b, (short)0, c,
                                               false, false);
}

__global__ void __launch_bounds__(128) gru_wmma_kernel(
    const float* __restrict__ x, const float* __restrict__ w_ih,
    const float* __restrict__ w_hh, const float* __restrict__ b_ih,
    const float* __restrict__ b_hh, const float* __restrict__ fc_w,
    const float* __restrict__ fc_b, float* __restrict__ out) {
  const int lane = threadIdx.x & 31;
  const int wid  = threadIdx.x >> 5;
  const int row  = lane & 15;
  const bool hi  = lane >= 16;       // upper half of wave: K+2/K+3, gate rows 8..15
  const int ka   = hi ? 2 : 0;
  const int bglob = blockIdx.x * 64 + wid * 16 + row;   // this lane-pair's batch

  // ---- preload weight A-operands (stay in VGPRs for the whole scan) ----
  const bool nrow = row < 8;
  const float* whr0 = w_hh + row * H_DIM;              // rows 0..15: r,z
  const float* whr1 = w_hh + (16 + (row & 7)) * H_DIM; // rows 16..23: n
  v2f a_hh_t0_k0 = mk2(whr0[ka],     whr0[ka + 1]);
  v2f a_hh_t0_k1 = mk2(whr0[4 + ka], whr0[5 + ka]);
  v2f a_hh_t1_k0 = nrow ? mk2(whr1[ka],     whr1[ka + 1]) : mk2(0.f, 0.f);
  v2f a_hh_t1_k1 = nrow ? mk2(whr1[4 + ka], whr1[5 + ka]) : mk2(0.f, 0.f);

  const float* wir0 = w_ih + row * I_DIM;
  const float* wir1 = w_ih + (16 + (row & 7)) * I_DIM;
  v2f a_ih_t0_k0 = mk2(wir0[ka], wir0[ka + 1]);                         // K 0..3
  v2f a_ih_t0_k1 = hi ? mk2(0.f, 0.f) : mk2(wir0[4], 0.f);              // K 4, pad 5..7
  v2f a_ih_t1_k0 = nrow ? mk2(wir1[ka], wir1[ka + 1]) : mk2(0.f, 0.f);
  v2f a_ih_t1_k1 = (nrow && !hi) ? mk2(wir1[4], 0.f) : mk2(0.f, 0.f);

  // ---- bias C seeds (D element j of this lane = gate row j + (hi?8:0)) ----
  v8f c_rz, c_xn, c_hn;
#pragma unroll
  for (int j = 0; j < 8; ++j) {
    int m = j + (hi ? 8 : 0);
    c_rz[j] = b_ih[m] + b_hh[m];            // r/z: biases fold together
    c_xn[j] = hi ? 0.f : b_ih[16 + j];      // n: b_ih outside r*(), b_hh inside
    c_hn[j] = hi ? 0.f : b_hh[16 + j];
  }

  // h in B-layout: lo lane holds h[0],h[1],h[4],h[5]; hi lane holds h[2],h[3],h[6],h[7]
  v2f h0 = mk2(0.f, 0.f), h1 = mk2(0.f, 0.f);

  const float* xb = x + (size_t)bglob * (T_DIM * I_DIM);
  float xa = xb[ka], xc = xb[ka + 1], xe = hi ? 0.f : xb[4];

  for (int t = 0; t < T_DIM; ++t) {
    // software pipeline: fetch next timestep's x while this step computes
    int tn = (t + 1 < T_DIM) ? t + 1 : t;
    const float* p = xb + tn * I_DIM;
    float nxa = p[ka], nxc = p[ka + 1], nxe = hi ? 0.f : p[4];
    __builtin_prefetch(xb + ((t + 16 < T_DIM) ? t + 16 : t) * I_DIM, 0, 1);

    v2f xB0 = mk2(xa, xc);        // K=ka..ka+1 of x (I padded to 8)
    v2f xB1 = mk2(xe, 0.f);       // K=4 on lo lanes, zeros elsewhere

    // r/z pre-activations: biases + W_ih*x + W_hh*h
    v8f rz = c_rz;
    rz = wmma4(a_ih_t0_k0, xB0, rz);
    rz = wmma4(a_ih_t0_k1, xB1, rz);
    rz = wmma4(a_hh_t0_k0, h0, rz);
    rz = wmma4(a_hh_t0_k1, h1, rz);
    // n pieces kept separate (torch: n = tanh(xn + b_in + r*(hn + b_hn)))
    v8f xn = c_xn;
    xn = wmma4(a_ih_t1_k0, xB0, xn);
    xn = wmma4(a_ih_t1_k1, xB1, xn);
    v8f hn = c_hn;
    hn = wmma4(a_hh_t1_k0, h0, hn);
    hn = wmma4(a_hh_t1_k1, h1, hn);

    float s[8], n[8];
#pragma unroll
    for (int j = 0; j < 8; ++j) s[j] = sigmoid_f(rz[j]);   // lo: r_j, hi: z_j
#pragma unroll
    for (int j = 0; j < 8; ++j) n[j] = tanh_hw(xn[j] + s[j] * hn[j]); // valid on lo

    // exchange exactly the halves each side needs (8 ds_swizzle / step)
    float zs0 = swz16(s[0]), zs1 = swz16(s[1]), zs4 = swz16(s[4]), zs5 = swz16(s[5]);
    float ns2 = swz16(n[2]), ns3 = swz16(n[3]), ns6 = swz16(n[6]), ns7 = swz16(n[7]);

    float z0 = hi ? s[2] : zs0, nn0 = hi ? ns2 : n[0];
    float z1 = hi ? s[3] : zs1, nn1 = hi ? ns3 : n[1];
    float z2 = hi ? s[6] : zs4, nn2 = hi ? ns6 : n[4];
    float z3 = hi ? s[7] : zs5, nn3 = hi ? ns7 : n[5];

    // h' = (1-z)*n + z*h = n + z*(h-n), directly in B-layout for next WMMA
    h0.x = nn0 + z0 * (h0.x - nn0);
    h0.y = nn1 + z1 * (h0.y - nn1);
    h1.x = nn2 + z2 * (h1.x - nn2);
    h1.y = nn3 + z3 * (h1.y - nn3);

    xa = nxa; xc = nxc; xe = nxe;
  }

  // ---- fused FC head: out = fc_w(2x8) @ h + fc_b; partner-swizzle reduce ----
  float o0 = fc_w[0 * 8 + ka] * h0.x + fc_w[0 * 8 + ka + 1] * h0.y +
             fc_w[0 * 8 + ka + 4] * h1.x + fc_w[0 * 8 + ka + 5] * h1.y;
  float o1 = fc_w[1 * 8 + ka] * h0.x + fc_w[1 * 8 + ka + 1] * h0.y +
             fc_w[1 * 8 + ka + 4] * h1.x + fc_w[1 * 8 + ka + 5] * h1.y;
  o0 += swz16(o0);
  o1 += swz16(o1);
  if (!hi) {
    out[(size_t)bglob * 2 + 0] = o0 + fc_b[0];
    out[(size_t)bglob * 2 + 1] = o1 + fc_b[1];
  }
}

extern "C" void kernel_launch(void* const* d_in, const int* in_sizes, int n_in,
                              void* d_out, int out_size, void* d_ws, size_t ws_size,
                              hipStream_t stream) {
  const float* x    = (const float*)d_in[0];
  const float* w_ih = (const float*)d_in[1];
  const float* w_hh = (const float*)d_in[2];
  const float* b_ih = (const float*)d_in[3];
  const float* b_hh = (const float*)d_in[4];
  const float* fc_w = (const float*)d_in[5];
  const float* fc_b = (const float*)d_in[6];
  float* out = (float*)d_out;

  const int B = in_sizes[0] / (T_DIM * I_DIM);   // 8192
  dim3 grid(B / 64), block(128);                 // 4 waves/block, 16 batches/wave
  gru_wmma_kernel<<<grid, block, 0, stream>>>(x, w_ih, w_hh, b_ih, b_hh, fc_w,
                                              fc_b, out);
}